// InteractionBlock_41523743818102
// MI455X (gfx1250) — compile-verified
//
#include <hip/hip_runtime.h>

typedef __attribute__((ext_vector_type(2))) float v2f;
typedef __attribute__((ext_vector_type(8))) float v8f;

#define EMB 128
#define NR  6
#define NS  7
#define NB  8
#define NCOMB (EMB + NB * EMB)   // 1152 combined output columns

// ---------------------------------------------------------------------------
// Pack B into a K-paired layout so each lane's (k, k+1) B pair is contiguous:
//   logical B[j, n]  ->  Bp[((j>>1)*NCOMB + n)*2 + (j&1)]
// Logical content:  cols 0..127          = W_m[j, n]
//                   cols 128 + b*128 + i = W_bilin[i, b, j]
// ---------------------------------------------------------------------------
__global__ void pack_B(const float* __restrict__ W_m,
                       const float* __restrict__ W_bilin,
                       float* __restrict__ Bp) {
  int idx = blockIdx.x * blockDim.x + threadIdx.x;
  if (idx >= EMB * NCOMB) return;
  int j = idx / NCOMB;
  int n = idx % NCOMB;
  float v;
  if (n < EMB) {
    v = W_m[j * EMB + n];
  } else {
    int b = (n - EMB) >> 7;
    int i = (n - EMB) & 127;
    v = W_bilin[(size_t)i * (NB * EMB) + b * EMB + j];
  }
  Bp[((size_t)(j >> 1) * NCOMB + n) * 2 + (j & 1)] = v;
}

__global__ void zero_f32(float* __restrict__ p, size_t n) {
  size_t i = (size_t)blockIdx.x * blockDim.x + threadIdx.x;
  size_t stride = (size_t)gridDim.x * blockDim.x;
  for (; i < n; i += stride) p[i] = 0.0f;
}

// ---------------------------------------------------------------------------
// GEMM: C[E,1152] = m[E,128] x B[128,1152] with V_WMMA_F32_16X16X4_F32.
// One wave computes a 16(M) x 64(N) tile (4 accumulators).
// Epilogue: cols < 128 -> w = (rbf@W_rbf) * silu(acc + b_m)  (into d_out)
//           cols >= 128 -> M workspace [E, 8*128]
// ---------------------------------------------------------------------------
__global__ __launch_bounds__(256) void gemm_edge(
    const float* __restrict__ A,      // m [E,128]
    const float* __restrict__ Bp,     // K-paired packed B [128,1152]
    const float* __restrict__ rbf,    // [E,6]
    const float* __restrict__ W_rbf,  // [6,128]
    const float* __restrict__ b_m,    // [128]
    float* __restrict__ w_out,        // [E,128]
    float* __restrict__ Mws,          // [E,1024]
    int E) {
  const int lane = threadIdx.x & 31;
  const int n_tiles = NCOMB / 64;            // 18
  const int m_tiles = (E + 15) >> 4;
  long tile = (long)blockIdx.x * 8 + (threadIdx.x >> 5);
  if (tile >= (long)m_tiles * n_tiles) return;   // wave-uniform exit
  const int n0 = (int)(tile % n_tiles) * 64;
  const int m0 = (int)(tile / n_tiles) * 16;

  const int row  = m0 + (lane & 15);         // A: lane -> M row
  const int kb   = (lane >> 4) * 2;          // lanes 16-31 hold K=2,3
  const int colb = n0 + (lane & 15);         // B: lane -> N column
  // Out-of-range rows (only possible in the last m-tile) would contribute to
  // C rows >= E, which the epilogue never stores -> just clamp the address.
  const int rowc = row < E ? row : (E - 1);
  const v2f* Arow = (const v2f*)(A + (size_t)rowc * EMB + kb);  // (k,k+1) pair

  v8f acc0 = {}, acc1 = {}, acc2 = {}, acc3 = {};

  for (int k = 0; k < EMB; k += 4) {
    const v2f a = Arow[k >> 1];                           // global_load_b64
    const int kpair = (k >> 1) + (lane >> 4);             // (k+kb)/2
    const v2f* Bk = (const v2f*)Bp + (size_t)kpair * NCOMB + colb;
    const v2f b0 = Bk[0];                                 // 4x global_load_b64
    const v2f b1 = Bk[16];
    const v2f b2 = Bk[32];
    const v2f b3 = Bk[48];
    acc0 = __builtin_amdgcn_wmma_f32_16x16x4_f32(false, a, false, b0, (short)0, acc0, false, false);
    acc1 = __builtin_amdgcn_wmma_f32_16x16x4_f32(false, a, false, b1, (short)0, acc1, false, false);
    acc2 = __builtin_amdgcn_wmma_f32_16x16x4_f32(false, a, false, b2, (short)0, acc2, false, false);
    acc3 = __builtin_amdgcn_wmma_f32_16x16x4_f32(false, a, false, b3, (short)0, acc3, false, false);
  }

  v8f accs[4] = {acc0, acc1, acc2, acc3};
  const int rowBase = m0 + ((lane >> 4) << 3);  // lanes 16-31 -> M+8
  const int nc = lane & 15;
#pragma unroll
  for (int t = 0; t < 4; ++t) {
    const int col = n0 + t * 16 + nc;
#pragma unroll
    for (int r = 0; r < 8; ++r) {
      const int orow = rowBase + r;
      if (orow >= E) continue;
      const float v = accs[t][r];
      if (col < EMB) {
        const float x = v + b_m[col];
        const float s = x / (1.0f + __expf(-x));   // silu
        float rw = 0.0f;
        const float* rp = rbf + (size_t)orow * NR;
#pragma unroll
        for (int q = 0; q < NR; ++q) rw += rp[q] * W_rbf[q * EMB + col];
        w_out[(size_t)orow * EMB + col] = rw * s;
      } else {
        Mws[(size_t)orow * (NB * EMB) + (col - EMB)] = v;
      }
    }
  }
}

// Guaranteed hardware f32 atomic add (no-return form, STOREcnt-tracked).
__device__ __forceinline__ void atomic_add_f32_hw(float* p, float v) {
  asm volatile("global_atomic_add_f32 %0, %1, off"
               :
               : "v"(p), "v"(v)
               : "memory");
}

// ---------------------------------------------------------------------------
// Triplet kernel: one wave32 per line-graph edge.
//   angle -> cos(s*theta) via Chebyshev; sbf[b] per-lane (b = lane&7);
//   x = sum_b sbf[b] * M[src,b,:]; atomic scatter into m_update[dst,:].
// ---------------------------------------------------------------------------
__global__ __launch_bounds__(256) void triplet_kernel(
    const float* __restrict__ rbf, const float* __restrict__ o,
    const int* __restrict__ lg_src, const int* __restrict__ lg_dst,
    const float* __restrict__ W_sbf, const float* __restrict__ Mws,
    float* __restrict__ out2, int L) {
  const int lane = threadIdx.x & 31;
  long l = (long)blockIdx.x * 8 + (threadIdx.x >> 5);
  if (l >= L) return;                         // wave-uniform exit
  const int src = lg_src[l];
  const int dst = lg_dst[l];

  const float* O1 = o + (size_t)src * 3;
  const float* O2 = o + (size_t)dst * 3;
  const float x1 = O1[0], y1 = O1[1], z1 = O1[2];
  const float x2 = O2[0], y2 = O2[1], z2 = O2[2];
  const float dotv = x1 * x2 + y1 * y2 + z1 * z2;
  const float cx = y1 * z2 - z1 * y2;
  const float cy = z1 * x2 - x1 * z2;
  const float cz = x1 * y2 - y1 * x2;
  const float d2 = dotv * dotv + cx * cx + cy * cy + cz * cz;
  const float ct = d2 > 0.0f ? dotv * rsqrtf(d2) : 1.0f;  // cos(theta)

  float c[NS];
  c[0] = 1.0f;
  c[1] = ct;
#pragma unroll
  for (int s = 2; s < NS; ++s) c[s] = 2.0f * ct * c[s - 1] - c[s - 2];

  // sbf[b] for b = lane & 7 (lanes replicate; wave32 shuffle broadcasts later)
  const int b = lane & 7;
  const float* rp = rbf + (size_t)src * NR;
  float rl[NR];
#pragma unroll
  for (int q = 0; q < NR; ++q) rl[q] = rp[q];
  float sbfv = 0.0f;
#pragma unroll
  for (int s = 0; s < NS; ++s) {
    float t = 0.0f;
#pragma unroll
    for (int q = 0; q < NR; ++q) t += rl[q] * W_sbf[(s * NR + q) * NB + b];
    sbfv += c[s] * t;
  }

  // x = sum_b sbf[b] * M[src,b,:]; lane handles 4 channels (float4 b128 loads)
  const float4* Mrow = (const float4*)(Mws + (size_t)src * (NB * EMB));
  float4 acc = make_float4(0.f, 0.f, 0.f, 0.f);
#pragma unroll
  for (int bb = 0; bb < NB; ++bb) {
    const float sb = __shfl(sbfv, bb, 32);
    const float4 mv = Mrow[bb * 32 + lane];
    acc.x += sb * mv.x; acc.y += sb * mv.y;
    acc.z += sb * mv.z; acc.w += sb * mv.w;
  }

  float* op = out2 + (size_t)dst * EMB + lane * 4;
  atomic_add_f32_hw(op + 0, acc.x);
  atomic_add_f32_hw(op + 1, acc.y);
  atomic_add_f32_hw(op + 2, acc.z);
  atomic_add_f32_hw(op + 3, acc.w);
}

// ---------------------------------------------------------------------------
extern "C" void kernel_launch(void* const* d_in, const int* in_sizes, int n_in,
                              void* d_out, int out_size, void* d_ws, size_t ws_size,
                              hipStream_t stream) {
  const float* rbf     = (const float*)d_in[0];
  const float* m       = (const float*)d_in[1];
  const float* o       = (const float*)d_in[2];
  const int*   lg_src  = (const int*)d_in[3];
  const int*   lg_dst  = (const int*)d_in[4];
  const float* W_rbf   = (const float*)d_in[5];
  const float* W_m     = (const float*)d_in[6];
  const float* b_m     = (const float*)d_in[7];
  const float* W_sbf   = (const float*)d_in[8];
  const float* W_bilin = (const float*)d_in[9];

  const int E = in_sizes[1] / EMB;
  const int L = in_sizes[3];

  float* w_out = (float*)d_out;                 // [E,128]
  float* out2  = w_out + (size_t)E * EMB;       // m_update [E,128]

  float* Bp  = (float*)d_ws;                    // 128*1152 floats (576 KB)
  float* Mws = Bp + (size_t)EMB * NCOMB;        // E*1024 floats (~410 MB)

  pack_B<<<(EMB * NCOMB + 255) / 256, 256, 0, stream>>>(W_m, W_bilin, Bp);
  zero_f32<<<2048, 256, 0, stream>>>(out2, (size_t)E * EMB);

  const long tiles = (long)((E + 15) / 16) * (NCOMB / 64);
  gemm_edge<<<(int)((tiles + 7) / 8), 256, 0, stream>>>(m, Bp, rbf, W_rbf, b_m,
                                                        w_out, Mws, E);

  triplet_kernel<<<(int)(((long)L + 7) / 8), 256, 0, stream>>>(
      rbf, o, lg_src, lg_dst, W_sbf, Mws, out2, L);
}